// FlowMIL_multi_tubes_13838384628105
// MI455X (gfx1250) — compile-verified
//
#include <hip/hip_runtime.h>

typedef __attribute__((ext_vector_type(2))) float v2f;
typedef __attribute__((ext_vector_type(8))) float v8f;

#define D_FEAT   32
#define N_HEADS  4
#define N_BAGS   64
#define FWS_ELEMS (N_BAGS * N_HEADS * D_FEAT)   // 8192
#define WSUM_ELEMS (N_BAGS * N_HEADS)           // 256
#define WAVES_TOTAL 4096
#define WAVES_PER_BLOCK 8                        // 256 threads / wave32

// ---------------------------------------------------------------- zero init
__global__ void flowmil_zero(float* __restrict__ ws, int n) {
  int i = blockIdx.x * blockDim.x + threadIdx.x;
  if (i < n) ws[i] = 0.0f;
}

// ---------------------------------------------------------------- accumulate
// fws[b][h][d] += att[i][h] * feat[i][d]  for seg[i]==b   (WMMA f32 16x16x4)
// wsum[b][h]   += att[i][h]
__global__ __launch_bounds__(256) void flowmil_accum(
    const float* __restrict__ feat,   // [n][32]
    const float* __restrict__ att,    // [n][4]
    const int*   __restrict__ seg2,   // int64 viewed as int pairs (low dword)
    float* __restrict__ fws,          // [64][4][32]
    float* __restrict__ wsum,         // [64][4]
    int n_rows, int rows_per_wave)
{
  const int lane = threadIdx.x & 31;
  // force wave-uniformity so seg loads become s_load + s_cmp/s_cbranch
  const int waveInBlk = __builtin_amdgcn_readfirstlane((int)(threadIdx.x >> 5));
  const int wave = blockIdx.x * WAVES_PER_BLOCK + waveInBlk;

  int rowStart = wave * rows_per_wave;
  int rowEnd   = rowStart + rows_per_wave;
  if (rowEnd > n_rows) rowEnd = n_rows;
  if (rowStart >= rowEnd) return;

  const int  m      = lane & 15;   // A row (head) / B-N column (feature)
  const int  khalf  = lane >> 4;   // 0 -> K{0,1}, 1 -> K{2,3}
  const int  hm     = m & 3;       // clamped head index (garbage rows unused)
  const bool isHead = (m < N_HEADS);

  const v8f vzero = {};
  v8f c0 = {};        // features 0..15
  v8f c1 = {};        // features 16..31
  float wacc = 0.0f;  // attention-sum partial (only head lanes flushed)

  int cur = seg2[2 * rowStart];    // uniform -> SMEM

  auto flush = [&](int bag) {
    if (lane < 16) {
#pragma unroll
      for (int v = 0; v < N_HEADS; ++v) {
        atomicAdd(&fws[bag * (N_HEADS * D_FEAT) + v * D_FEAT + m],       c0[v]);
        atomicAdd(&fws[bag * (N_HEADS * D_FEAT) + v * D_FEAT + m + 16],  c1[v]);
      }
    }
    if (isHead) atomicAdd(&wsum[bag * N_HEADS + m], wacc);
    c0 = vzero; c1 = vzero; wacc = 0.0f;
  };

  // ---- hot path: 4 rows, bag-uniform, all in range, no masks ----------
  auto fastGroup = [&](int g) {
    const int rA = g + 2 * khalf;
    const int rB = rA + 1;
    v2f a, b0, b1;
    a.x  = att[rA * N_HEADS + hm];       a.y  = att[rB * N_HEADS + hm];
    b0.x = feat[rA * D_FEAT + m];        b0.y = feat[rB * D_FEAT + m];
    b1.x = feat[rA * D_FEAT + m + 16];   b1.y = feat[rB * D_FEAT + m + 16];
    wacc += a.x + a.y;
    c0 = __builtin_amdgcn_wmma_f32_16x16x4_f32(false, a, false, b0, (short)0, c0, false, false);
    c1 = __builtin_amdgcn_wmma_f32_16x16x4_f32(false, a, false, b1, (short)0, c1, false, false);
  };

  // ---- generic path: 4 rows with bag masks + range clamps (rare) ------
  auto slowGroup = [&](int g) {
    const int e  = rowEnd - 1;
    const int i1 = (g + 1 < rowEnd) ? g + 1 : e;
    const int i2 = (g + 2 < rowEnd) ? g + 2 : e;
    const int i3 = (g + 3 < rowEnd) ? g + 3 : e;
    const int s0 = seg2[2 * g];          // uniform -> SMEM
    const int s1 = seg2[2 * i1];
    const int s2 = seg2[2 * i2];
    const int s3 = seg2[2 * i3];

    const int  rA  = g + 2 * khalf;
    const int  rB  = rA + 1;
    const int  rAc = (rA < rowEnd) ? rA : e;
    const int  rBc = (rB < rowEnd) ? rB : e;
    const int  sA  = khalf ? s2 : s0;
    const int  sB  = khalf ? s3 : s1;
    const bool vA  = (rA < rowEnd);
    const bool vB  = (rB < rowEnd);

    const float attA = att[rAc * N_HEADS + hm];
    const float attB = att[rBc * N_HEADS + hm];
    const float f0x  = feat[rAc * D_FEAT + m];      const float f0y = feat[rBc * D_FEAT + m];
    const float f1x  = feat[rAc * D_FEAT + m + 16]; const float f1y = feat[rBc * D_FEAT + m + 16];

    while (true) {
      const float mA = (vA && sA == cur) ? 1.0f : 0.0f;
      const float mB = (vB && sB == cur) ? 1.0f : 0.0f;
      v2f a, b0, b1;
      a.x = attA * mA; a.y = attB * mB;
      b0.x = f0x; b0.y = f0y;
      b1.x = f1x; b1.y = f1y;
      wacc += a.x + a.y;
      c0 = __builtin_amdgcn_wmma_f32_16x16x4_f32(false, a, false, b0, (short)0, c0, false, false);
      c1 = __builtin_amdgcn_wmma_f32_16x16x4_f32(false, a, false, b1, (short)0, c1, false, false);
      if (s3 == cur) break;             // sorted: last valid row handled
      flush(cur);
      cur = (s0 > cur) ? s0 : (s1 > cur) ? s1 : (s2 > cur) ? s2 : s3;
    }
  };

  // ---- main loop: 8 rows / iteration ---------------------------------
  int r0 = rowStart;
  const int fullEnd = rowStart + ((rowEnd - rowStart) & ~7);
  for (; r0 < fullEnd; r0 += 8) {
    const int s0 = seg2[2 * r0];        // uniform -> SMEM
    const int s7 = seg2[2 * (r0 + 7)];
    if ((s0 == cur) & (s7 == cur)) {    // uniform -> s_cbranch, no divergence
      fastGroup(r0);
      fastGroup(r0 + 4);
    } else {
      slowGroup(r0);
      slowGroup(r0 + 4);
    }
  }
  for (; r0 < rowEnd; r0 += 4) slowGroup(r0);   // tail (<8 rows)

  flush(cur);
}

// ---------------------------------------------------------------- finalize
__global__ void flowmil_finalize(const float* __restrict__ fws,
                                 const float* __restrict__ wsum,
                                 float* __restrict__ out)
{
  int i = blockIdx.x * blockDim.x + threadIdx.x;
  if (i < FWS_ELEMS) {
    const int b = i >> 7;            // /(4*32)
    const int h = (i >> 5) & 3;
    const float den = wsum[b * N_HEADS + h];
    float v = (den == 0.0f) ? 0.0f : fws[i] / den;
    if (v != v) v = 1e-05f;          // NaN -> 1e-5
    out[i] = v;
  }
  if (i < WSUM_ELEMS) out[FWS_ELEMS + i] = wsum[i];
}

// ---------------------------------------------------------------- launch
extern "C" void kernel_launch(void* const* d_in, const int* in_sizes, int n_in,
                              void* d_out, int out_size, void* d_ws, size_t ws_size,
                              hipStream_t stream) {
  (void)n_in; (void)out_size; (void)ws_size;
  const float* feat = (const float*)d_in[0];
  const float* att  = (const float*)d_in[1];
  const int*   seg2 = (const int*)d_in[2];   // int64 -> read low dwords
  const int n_rows  = in_sizes[0] / D_FEAT;

  float* fws  = (float*)d_ws;                // 8192 f32
  float* wsum = fws + FWS_ELEMS;             // 256 f32

  flowmil_zero<<<(FWS_ELEMS + WSUM_ELEMS + 255) / 256, 256, 0, stream>>>(
      fws, FWS_ELEMS + WSUM_ELEMS);

  int rpw = ((n_rows + WAVES_TOTAL * 8 - 1) / (WAVES_TOTAL * 8)) * 8;
  if (rpw < 8) rpw = 8;
  int blocks = (WAVES_TOTAL + WAVES_PER_BLOCK - 1) / WAVES_PER_BLOCK;
  flowmil_accum<<<blocks, WAVES_PER_BLOCK * 32, 0, stream>>>(
      feat, att, seg2, fws, wsum, n_rows, rpw);

  flowmil_finalize<<<(FWS_ELEMS + 255) / 256, 256, 0, stream>>>(fws, wsum, (float*)d_out);
}